// TopologyEncoder_25039659336365
// MI455X (gfx1250) — compile-verified
//
#include <hip/hip_runtime.h>

// CDNA5 / gfx1250 implementation of a 2-layer GCN + pool + MLP.
// f32 WMMA (V_WMMA_F32_16X16X4_F32) used for the dense GEMMs.

typedef float v2f __attribute__((ext_vector_type(2)));
typedef float v8f __attribute__((ext_vector_type(8)));

#define NB 64      // graphs per batch
#define F 32       // hidden feature width

__global__ void zero_f32(float* __restrict__ p, int n) {
  int i = blockIdx.x * blockDim.x + threadIdx.x;
  if (i < n) p[i] = 0.f;
}

// deg[dst] += 1 per edge (self-loop "+1" folded into make_dinv)
__global__ void deg_count(const int* __restrict__ dst, int E, float* __restrict__ deg) {
  int e = blockIdx.x * blockDim.x + threadIdx.x;
  if (e < E) atomicAdd(&deg[dst[e]], 1.0f);
}

__global__ void make_dinv(float* __restrict__ deg, int N) {
  int n = blockIdx.x * blockDim.x + threadIdx.x;
  if (n < N) deg[n] = rsqrtf(deg[n] + 1.0f);   // deg+selfloop >= 1 always
}

// hx[n,f] = sum_{k<3} x[n,k] * W1[k,f]   (K=3: scalar is the right tool)
__global__ void xw1_kernel(const float* __restrict__ x, const float* __restrict__ W1,
                           float* __restrict__ hx, int N) {
  int i = blockIdx.x * blockDim.x + threadIdx.x;
  if (i >= N * F) return;
  int n = i >> 5, f = i & 31;
  const float* xr = x + n * 3;
  hx[i] = xr[0] * W1[f] + xr[1] * W1[F + f] + xr[2] * W1[2 * F + f];
}

// One 32-lane group per edge: lane = feature. 128B coalesced gather + 128B L2 atomic add.
__global__ void scatter_edges(const int* __restrict__ src, const int* __restrict__ dst,
                              const float* __restrict__ dinv, const float* __restrict__ hin,
                              float* __restrict__ acc, int E) {
  int e = blockIdx.x * 8 + (threadIdx.x >> 5);
  if (e >= E) return;
  int f = threadIdx.x & 31;
  int s = src[e], d = dst[e];
  float nrm = dinv[s] * dinv[d];                 // same addr across lanes -> broadcast
  float v = nrm * hin[s * F + f];                // coalesced 128B per edge-group
  atomicAdd(&acc[d * F + f], v);                 // GLOBAL_ATOMIC_ADD_F32, resolves in L2
}

// h = relu(acc + dinv^2 * hself + bias); optional store; LDS-staged segment max/sum/cnt.
__global__ void finalize_pool(const float* __restrict__ acc, const float* __restrict__ hself,
                              const float* __restrict__ dinv, const float* __restrict__ bias,
                              const int* __restrict__ batch,
                              float* __restrict__ hout,                     // nullptr in layer 2
                              float* __restrict__ gmax, float* __restrict__ gsum,
                              float* __restrict__ gcnt,                     // nullptr in layer 2
                              int N) {
  __shared__ float smax[NB * F];
  __shared__ float ssum[NB * F];
  __shared__ float scnt[NB];
  for (int i = threadIdx.x; i < NB * F; i += blockDim.x) { smax[i] = 0.f; ssum[i] = 0.f; }
  if (threadIdx.x < NB) scnt[threadIdx.x] = 0.f;
  __syncthreads();

  int f = threadIdx.x & 31;
  int grp = threadIdx.x >> 5;          // 8 node-groups per 256-thread block
  int base = blockIdx.x * 256;
  for (int r = 0; r < 32; ++r) {
    int n = base + r * 8 + grp;
    if (n < N) {
      float di = dinv[n];
      float h = acc[n * F + f] + di * di * hself[n * F + f] + bias[f];
      h = fmaxf(h, 0.f);
      if (hout) hout[n * F + f] = h;
      int b = batch[n];
      atomicMax((int*)&smax[b * F + f], __float_as_int(h));  // valid: h >= 0
      atomicAdd(&ssum[b * F + f], h);
      if (f == 0 && gcnt) atomicAdd(&scnt[b], 1.f);
    }
  }
  __syncthreads();
  // batch is sorted: each block touches ~1-2 segments -> few global atomics
  for (int i = threadIdx.x; i < NB * F; i += blockDim.x) {
    if (smax[i] != 0.f || ssum[i] != 0.f) {
      atomicMax((int*)&gmax[i], __float_as_int(smax[i]));
      atomicAdd(&gsum[i], ssum[i]);
    }
  }
  if (gcnt && threadIdx.x < NB && scnt[threadIdx.x] != 0.f)
    atomicAdd(&gcnt[threadIdx.x], scnt[threadIdx.x]);
}

// hx2 = H @ W2 : [N,32] @ [32,32] via V_WMMA_F32_16X16X4_F32.
// One wave per 16-node tile; 8 K-steps x 2 N-tiles = 16 WMMAs/wave; full EXEC throughout.
__global__ void gemm_h_w2(const float* __restrict__ H, const float* __restrict__ W2,
                          float* __restrict__ out, int N) {
  int lane = threadIdx.x & 31;
  int wave = threadIdx.x >> 5;
  int tile = blockIdx.x * 8 + wave;
  int row0 = tile * 16;
  int m = lane & 15;                      // A-frag row (M)
  int kh = (lane >> 4) << 1;              // K pair: lanes 0-15 -> K+0/1, 16-31 -> K+2/3
  int node = row0 + m; if (node >= N) node = N - 1;   // clamp load, predicate store
  const float* hr = H + node * F;
  int nn = lane & 15;                     // B/D column (N)
  v8f c0 = {0,0,0,0,0,0,0,0}, c1 = {0,0,0,0,0,0,0,0};
  for (int k = 0; k < F; k += 4) {
    v2f a, b0, b1;
    a.x = hr[k + kh];  a.y = hr[k + kh + 1];
    const float* w0 = W2 + (k + kh) * F;  // W2 row-major [32,32]
    b0.x = w0[nn];       b0.y = w0[F + nn];
    b1.x = w0[16 + nn];  b1.y = w0[F + 16 + nn];
    c0 = __builtin_amdgcn_wmma_f32_16x16x4_f32(false, a, false, b0, (short)0, c0, false, false);
    c1 = __builtin_amdgcn_wmma_f32_16x16x4_f32(false, a, false, b1, (short)0, c1, false, false);
  }
  int mh = (lane >> 4) << 3;              // D: M = vgpr + 8*(lane>=16)
  for (int v = 0; v < 8; ++v) {
    int n2 = row0 + mh + v;
    if (n2 < N) {
      out[n2 * F + nn]      = c0[v];
      out[n2 * F + 16 + nn] = c1[v];
    }
  }
}

// Single-block MLP: P = x1+x2 (from pools), Z = relu(P@LW1+Lb1), out = Z@LW2+Lb2.
// 64x64x64 GEMMs, 16 output tiles each, f32 WMMA, 2 tiles per wave.
__global__ void mlp_kernel(const float* __restrict__ pool,
                           const float* __restrict__ LW1, const float* __restrict__ Lb1,
                           const float* __restrict__ LW2, const float* __restrict__ Lb2,
                           float* __restrict__ out) {
  __shared__ float P[64 * 64];
  __shared__ float Z[64 * 64];
  const float* maxs1 = pool;
  const float* sums1 = pool + 2048;
  const float* cnt   = pool + 4096;
  const float* maxs2 = pool + 4160;
  const float* sums2 = pool + 4160 + 2048;

  for (int i = threadIdx.x; i < 4096; i += 256) {
    int b = i >> 6, f = i & 63;
    float c = fmaxf(cnt[b], 1.f);
    float v1 = (f < 32) ? maxs1[b * 32 + f] : sums1[b * 32 + (f - 32)] / c;
    float v2 = (f < 32) ? maxs2[b * 32 + f] : sums2[b * 32 + (f - 32)] / c;
    P[i] = v1 + v2;
  }
  __syncthreads();

  int lane = threadIdx.x & 31;
  int wave = threadIdx.x >> 5;
  int m  = lane & 15;
  int kh = (lane >> 4) << 1;
  int nn = lane & 15;
  int mh = (lane >> 4) << 3;

  for (int t = wave; t < 16; t += 8) {            // GEMM1 -> Z (relu)
    int mt = (t >> 2) * 16, nt = (t & 3) * 16;
    v8f c = {0,0,0,0,0,0,0,0};
    for (int k = 0; k < 64; k += 4) {
      v2f a, b;
      a.x = P[(mt + m) * 64 + k + kh];        a.y = P[(mt + m) * 64 + k + kh + 1];
      b.x = LW1[(k + kh) * 64 + nt + nn];     b.y = LW1[(k + kh + 1) * 64 + nt + nn];
      c = __builtin_amdgcn_wmma_f32_16x16x4_f32(false, a, false, b, (short)0, c, false, false);
    }
    for (int v = 0; v < 8; ++v) {
      int row = mt + mh + v, col = nt + nn;
      Z[row * 64 + col] = fmaxf(c[v] + Lb1[col], 0.f);
    }
  }
  __syncthreads();

  for (int t = wave; t < 16; t += 8) {            // GEMM2 -> out
    int mt = (t >> 2) * 16, nt = (t & 3) * 16;
    v8f c = {0,0,0,0,0,0,0,0};
    for (int k = 0; k < 64; k += 4) {
      v2f a, b;
      a.x = Z[(mt + m) * 64 + k + kh];        a.y = Z[(mt + m) * 64 + k + kh + 1];
      b.x = LW2[(k + kh) * 64 + nt + nn];     b.y = LW2[(k + kh + 1) * 64 + nt + nn];
      c = __builtin_amdgcn_wmma_f32_16x16x4_f32(false, a, false, b, (short)0, c, false, false);
    }
    for (int v = 0; v < 8; ++v) {
      int row = mt + mh + v, col = nt + nn;
      out[row * 64 + col] = c[v] + Lb2[col];
    }
  }
}

extern "C" void kernel_launch(void* const* d_in, const int* in_sizes, int n_in,
                              void* d_out, int out_size, void* d_ws, size_t ws_size,
                              hipStream_t stream) {
  const float* x   = (const float*)d_in[0];
  const int*   ei  = (const int*)  d_in[1];
  const int*   bat = (const int*)  d_in[2];
  const float* W1  = (const float*)d_in[3];
  const float* b1  = (const float*)d_in[4];
  const float* W2  = (const float*)d_in[5];
  const float* b2  = (const float*)d_in[6];
  const float* LW1 = (const float*)d_in[7];
  const float* Lb1 = (const float*)d_in[8];
  const float* LW2 = (const float*)d_in[9];
  const float* Lb2 = (const float*)d_in[10];

  const int N = in_sizes[0] / 3;
  const int E = in_sizes[1] / 2;
  const int* src = ei;       // edge_index[0]
  const int* dst = ei + E;   // edge_index[1]

  // Workspace: dinv[N] | acc[32N] | hx[32N] (xW1, reused as hx2) | H[32N] | pools[8256]
  float* ws    = (float*)d_ws;
  float* dinv  = ws;
  float* acc   = dinv + N;
  float* hx    = acc + 32 * (size_t)N;
  float* H     = hx + 32 * (size_t)N;
  float* pools = H + 32 * (size_t)N;
  float* maxs1 = pools;
  float* sums1 = pools + 2048;
  float* cnt   = pools + 4096;
  float* maxs2 = pools + 4160;
  float* sums2 = pools + 4160 + 2048;

  auto cdiv = [](int a, int b) { return (a + b - 1) / b; };

  // init (harness does not re-zero between replays)
  zero_f32<<<cdiv(N, 256), 256, 0, stream>>>(dinv, N);
  zero_f32<<<cdiv(8256, 256), 256, 0, stream>>>(pools, 8256);
  zero_f32<<<cdiv(32 * N, 256), 256, 0, stream>>>(acc, 32 * N);

  // normalization
  deg_count<<<cdiv(E, 256), 256, 0, stream>>>(dst, E, dinv);
  make_dinv<<<cdiv(N, 256), 256, 0, stream>>>(dinv, N);

  // layer 1
  xw1_kernel<<<cdiv(32 * N, 256), 256, 0, stream>>>(x, W1, hx, N);
  scatter_edges<<<cdiv(E, 8), 256, 0, stream>>>(src, dst, dinv, hx, acc, E);
  finalize_pool<<<cdiv(N, 256), 256, 0, stream>>>(acc, hx, dinv, b1, bat,
                                                  H, maxs1, sums1, cnt, N);

  // layer 2
  zero_f32<<<cdiv(32 * N, 256), 256, 0, stream>>>(acc, 32 * N);
  gemm_h_w2<<<cdiv(N, 128), 256, 0, stream>>>(H, W2, hx, N);     // hx now hx2
  scatter_edges<<<cdiv(E, 8), 256, 0, stream>>>(src, dst, dinv, hx, acc, E);
  finalize_pool<<<cdiv(N, 256), 256, 0, stream>>>(acc, hx, dinv, b2, bat,
                                                  nullptr, maxs2, sums2, nullptr, N);

  // head
  mlp_kernel<<<1, 256, 0, stream>>>(pools, LW1, Lb1, LW2, Lb2, (float*)d_out);
}